// DePooling2D_64235530879511
// MI455X (gfx1250) — compile-verified
//
#include <hip/hip_runtime.h>

// Max-unpool (DePooling2D), stride 2, NHWC.
//   net  : [8,56,56,256]  f32   (pooled values)
//   mask : [8,56,56,256]  int   (flat argmax: ((y*oW)+x)*C + c, per-batch)
//   out  : [8,112,112,256] f32
//
// Each pooled element scatters into its private 2x2 window (disjoint by
// construction of the argmax encoding), so each thread writes the entire
// window: value at (dy,dx), zeros elsewhere. Single pass, no atomics, no
// separate zero-fill. Pure streaming workload -> non-temporal hints on all
// global traffic. Index decode is mapped onto the hardware grid so no
// integer division survives: grid = (W/4, H, B), block = 64 chan-quads x 4 w.

typedef __attribute__((ext_vector_type(4))) float f32x4;
typedef __attribute__((ext_vector_type(4))) int   i32x4;

namespace {
constexpr int kB  = 8;
constexpr int kH  = 56;
constexpr int kW  = 56;
constexpr int kC  = 256;
constexpr int kOH = kH * 2;               // 112
constexpr int kOW = kW * 2;               // 112
constexpr int kBlock = 256;               // 8 wave32s
constexpr int kWPerBlock = 4;             // w positions per block
}

__global__ __launch_bounds__(kBlock)
void depool_scatter_kernel(const float* __restrict__ net,
                           const int*   __restrict__ mask,
                           float*       __restrict__ out)
{
    // Lanes 0..63 cover channels (vec4), lanes advance c fastest -> coalesced.
    const unsigned c0 = (threadIdx.x & 63u) << 2;                  // 0..252
    const unsigned w  = (blockIdx.x << 2) + (threadIdx.x >> 6);    // 0..55
    const unsigned h  = blockIdx.y;                                // 0..55
    const unsigned b  = blockIdx.z;                                // 0..7

    // Flat pooled-element index (shifts/mads only).
    const unsigned base = (((b * (unsigned)kH + h) * (unsigned)kW + w) << 8) + c0;

    // Streamed 128-bit loads (read-once -> non-temporal).
    const f32x4 v = __builtin_nontemporal_load((const f32x4*)(net  + base));
    const i32x4 m = __builtin_nontemporal_load((const i32x4*)(mask + base));

    // mask>>8 == y*oW + x (C == 256). Subtract window origin -> dy*oW + dx.
    const unsigned rowoff = (2u * h) * (unsigned)kOW + 2u * w;

    f32x4 q00 = {0.f, 0.f, 0.f, 0.f};   // (dy=0,dx=0)
    f32x4 q01 = {0.f, 0.f, 0.f, 0.f};   // (dy=0,dx=1)
    f32x4 q10 = {0.f, 0.f, 0.f, 0.f};   // (dy=1,dx=0)
    f32x4 q11 = {0.f, 0.f, 0.f, 0.f};   // (dy=1,dx=1)

#pragma unroll
    for (int j = 0; j < 4; ++j) {
        const unsigned t   = (((unsigned)m[j]) >> 8) - rowoff;  // {0,1,112,113}
        const float    val = v[j];
        q00[j] = (t == 0u)                 ? val : 0.0f;
        q01[j] = (t == 1u)                 ? val : 0.0f;
        q10[j] = (t == (unsigned)kOW)      ? val : 0.0f;
        q11[j] = (t == (unsigned)kOW + 1u) ? val : 0.0f;
    }

    // Window origin in the output tensor (shifts/mads only).
    const unsigned obase =
        ((b * (unsigned)kOH + 2u * h) * (unsigned)kOW + 2u * w) * (unsigned)kC + c0;
    float* o = out + obase;

    // Four coalesced 128-bit streams per wave (512 B contiguous each);
    // write-once -> non-temporal.
    __builtin_nontemporal_store(q00, (f32x4*)(o));
    __builtin_nontemporal_store(q01, (f32x4*)(o + kC));
    __builtin_nontemporal_store(q10, (f32x4*)(o + (unsigned)kOW * kC));
    __builtin_nontemporal_store(q11, (f32x4*)(o + (unsigned)kOW * kC + kC));
}

extern "C" void kernel_launch(void* const* d_in, const int* in_sizes, int n_in,
                              void* d_out, int out_size, void* d_ws, size_t ws_size,
                              hipStream_t stream)
{
    const float* net  = (const float*)d_in[0];
    const int*   mask = (const int*)d_in[1];
    // d_in[2] is the stride scalar (== 2); baked in at compile time.
    float* out = (float*)d_out;

    dim3 grid(kW / kWPerBlock, kH, kB);   // (14, 56, 8)
    depool_scatter_kernel<<<grid, dim3(kBlock), 0, stream>>>(net, mask, out);
}